// TopKPathMoE_64965675319853
// MI455X (gfx1250) — compile-verified
//
#include <hip/hip_runtime.h>
#include <math.h>
#include <stdint.h>

// Problem constants (match reference)
#define Bsz  1024
#define Gg   2048
#define Pp   256
#define Dd   (3 * Gg)     // 6144
#define GHh  128
#define EHh  64
#define EOo  16
#define CHh  8
#define Kk   3
#define NCc  1
#define BN_EPS 1e-5f

#define CHD  32           // K-chunk (rows of eW1) staged per async panel

typedef __attribute__((ext_vector_type(2))) float v2f;
typedef __attribute__((ext_vector_type(8))) float v8f;
typedef __attribute__((__vector_size__(16))) int v4i;

#if __has_builtin(__builtin_amdgcn_global_load_async_to_lds_b128)
#define ASYNC_LDS 1
#else
#define ASYNC_LDS 0
#endif

// D = A(16x4) * B(4x16) + C, fp32 WMMA (CDNA5)
__device__ __forceinline__ v8f wmma4(v2f a, v2f b, v8f c) {
    return __builtin_amdgcn_wmma_f32_16x16x4_f32(false, a, false, b, (short)0, c, false, false);
}

// 16-byte global -> LDS copy (async on CDNA5, direct fallback otherwise)
__device__ __forceinline__ void copy16_g2l(const float* g, float* l) {
#if ASYNC_LDS
    __builtin_amdgcn_global_load_async_to_lds_b128(
        (__attribute__((address_space(1))) v4i*)(uintptr_t)g,
        (__attribute__((address_space(3))) v4i*)(unsigned)(uintptr_t)l,
        0, 0);
#else
    *(float4*)l = *(const float4*)g;
#endif
}

// Wait for staged panel: with a successor panel in flight keep 4 async ops
// outstanding, otherwise drain. Compiler barrier pins LDS reads behind it.
__device__ __forceinline__ void wait_panel(bool more_in_flight) {
#if ASYNC_LDS
#if __has_builtin(__builtin_amdgcn_s_wait_asynccnt)
    if (more_in_flight) __builtin_amdgcn_s_wait_asynccnt(4);
    else                __builtin_amdgcn_s_wait_asynccnt(0);
#else
    if (more_in_flight) asm volatile("s_wait_asynccnt 0x4" ::: "memory");
    else                asm volatile("s_wait_asynccnt 0x0" ::: "memory");
#endif
#endif
    asm volatile("" ::: "memory");
}

// Stage one 32-row x 16-col fp32 panel of eW1 (2KB) into this wave's LDS slot.
// Lane l copies row l (16 floats = 4 x b128).
__device__ __forceinline__ void stage_panel(const float* w1col, float* dst, int lane) {
    const float* g = w1col + (size_t)lane * EHh;   // row stride = 64 floats
    float*       l = dst + lane * 16;
    copy16_g2l(g + 0,  l + 0);
    copy16_g2l(g + 4,  l + 4);
    copy16_g2l(g + 8,  l + 8);
    copy16_g2l(g + 12, l + 12);
}

// ---------------------------------------------------------------------------
// Kernel 0: zero per-launch scratch (bucket counts + weighted accumulator)
// ---------------------------------------------------------------------------
__global__ void k_zero(int* __restrict__ counts, float* __restrict__ weighted) {
    int i = blockIdx.x * blockDim.x + threadIdx.x;
    if (i < Pp) counts[i] = 0;
    if (i < Bsz * EOo) weighted[i] = 0.f;
}

// ---------------------------------------------------------------------------
// Kernel 1: gh = relu(x_rna @ gate_W1 + b1)   [1024x2048]x[2048x128]
// One wave per 16x16 output tile, WMMA f32 16x16x4, K step 4.
// ---------------------------------------------------------------------------
__global__ void k_gate_hidden(const float* __restrict__ x,
                              const float* __restrict__ W1,
                              const float* __restrict__ b1,
                              float* __restrict__ gh) {
    const int row0 = blockIdx.x * 16;
    const int col0 = blockIdx.y * 16;
    const int lane = threadIdx.x & 31;
    const int half = lane >> 4;     // 0 or 1
    const int lm   = lane & 15;     // M for A, N for B/C/D
    const int ka   = half * 2;      // this lane's K base within a 4-step

    const float* __restrict__ xr = x + (size_t)(row0 + lm) * Gg;
    v8f c = {};
#pragma unroll 4
    for (int k = 0; k < Gg; k += 4) {
        v2f a; a.x = xr[k + ka]; a.y = xr[k + ka + 1];                 // contiguous pair
        v2f b; b.x = W1[(size_t)(k + ka) * GHh + col0 + lm];
               b.y = W1[(size_t)(k + ka + 1) * GHh + col0 + lm];
        c = wmma4(a, b, c);
    }
    const float bias = b1[col0 + lm];
#pragma unroll
    for (int j = 0; j < 8; ++j) {
        float h = fmaxf(c[j] + bias, 0.f);
        gh[(size_t)(row0 + j + 8 * half) * GHh + col0 + lm] = h;
    }
}

// ---------------------------------------------------------------------------
// Kernel 2: per-row gate logits, top-3, sigmoid weights.
// Writes gate_weights output rows and scatters (row, weight) into per-expert
// buckets for the grouped expert kernel.
// ---------------------------------------------------------------------------
__global__ void k_gate_topk(const float* __restrict__ gh,
                            const float* __restrict__ W2,
                            const float* __restrict__ b2,
                            float* __restrict__ gw_out,        // [B, P] portion of d_out
                            int*   __restrict__ counts,
                            int*   __restrict__ rows,          // [P, B]
                            float* __restrict__ wvals) {       // [P, B]
    const int b = blockIdx.x;
    const int t = threadIdx.x;              // 0..255, one per pathway

    __shared__ float s_gh[GHh];
    __shared__ float s_log[Pp];
    __shared__ float s_v[Pp];
    __shared__ int   s_i[Pp];
    __shared__ int   topi[Kk];
    __shared__ float topv[Kk];

    if (t < GHh) s_gh[t] = gh[(size_t)b * GHh + t];
    __syncthreads();

    float acc = b2[t];
#pragma unroll 8
    for (int k = 0; k < GHh; ++k)
        acc += s_gh[k] * W2[(size_t)k * Pp + t];
    s_log[t] = acc;
    gw_out[(size_t)b * Pp + t] = 0.f;       // zero the whole row first
    __syncthreads();

    // 3 rounds of parallel argmax (ties -> lowest index, matching lax.top_k)
    for (int kk = 0; kk < Kk; ++kk) {
        s_v[t] = s_log[t]; s_i[t] = t;
        __syncthreads();
        for (int s = Pp / 2; s > 0; s >>= 1) {
            if (t < s) {
                bool take = (s_v[t + s] > s_v[t]) ||
                            (s_v[t + s] == s_v[t] && s_i[t + s] < s_i[t]);
                if (take) { s_v[t] = s_v[t + s]; s_i[t] = s_i[t + s]; }
            }
            __syncthreads();
        }
        if (t == 0) {
            topi[kk] = s_i[0]; topv[kk] = s_v[0];
            s_log[s_i[0]] = -INFINITY;
        }
        __syncthreads();
    }

    if (t < Kk) {
        const int   idx = topi[t];
        const float w   = 1.f / (1.f + expf(-topv[t]));   // sigmoid
        gw_out[(size_t)b * Pp + idx] = w;
        const int slot = atomicAdd(&counts[idx], 1);
        rows [idx * Bsz + slot] = b;
        wvals[idx * Bsz + slot] = w;
    }
}

// ---------------------------------------------------------------------------
// Kernel 3: grouped sparse experts. One block (4 waves) per expert.
// Each wave owns 16 of the 64 hidden columns for a 16-row tile of selected
// batch rows. The bandwidth-critical eW1 stream is double-buffered into LDS
// with GLOBAL_LOAD_ASYNC_TO_LDS_B128 (ASYNCcnt-synchronized), overlapping the
// next 2KB B-panel fetch with WMMA on the current one. Mask is folded into A.
// Then BN+ReLU, stage h in LDS, wave 0 runs the 64->16 eW2 GEMM with WMMA and
// atomically accumulates weight * feat into `weighted`.
// ---------------------------------------------------------------------------
__global__ void k_expert(const float* __restrict__ x_rna,
                         const float* __restrict__ x_cnv,
                         const float* __restrict__ x_met,
                         const float* __restrict__ gene_mask,   // [G, P]
                         const float* __restrict__ eW1,         // [P, D, EH]
                         const float* __restrict__ eb1,         // [P, EH]
                         const float* __restrict__ bn_g,
                         const float* __restrict__ bn_b,
                         const float* __restrict__ bn_m,
                         const float* __restrict__ bn_v,
                         const float* __restrict__ eW2,         // [P, EH, EO]
                         const float* __restrict__ eb2,         // [P, EO]
                         const int*   __restrict__ counts,
                         const int*   __restrict__ rows,
                         const float* __restrict__ wvals,
                         float*       __restrict__ weighted) {  // [B, EO]
    const int p    = blockIdx.x;
    const int wave = threadIdx.x >> 5;
    const int lane = threadIdx.x & 31;
    const int half = lane >> 4;
    const int lm   = lane & 15;
    const int ka   = half * 2;

    const int n = counts[p];
    if (n == 0) return;

    __shared__ int   s_rows[16];
    __shared__ float s_w[16];
    __shared__ float s_h[16 * EHh];            // 16 rows x 64 cols of hidden acts
    __shared__ float s_b[2][4][CHD * 16];      // double-buffered B panels, 16KB

    // base of this wave's 16-column strip of eW1[p]
    const float* w1base = eW1 + (size_t)p * Dd * EHh + wave * 16;

    for (int t0 = 0; t0 < n; t0 += 16) {
        if (threadIdx.x < 16) {
            int j = t0 + threadIdx.x;
            s_rows[threadIdx.x] = (j < n) ? rows [p * Bsz + j] : 0;
            s_w  [threadIdx.x]  = (j < n) ? wvals[p * Bsz + j] : 0.f;
        }
        __syncthreads();

        const int brow = s_rows[lm];            // A: M = lm for both halves
        v8f c = {};

        // prologue: stage panel 0
        stage_panel(w1base, &s_b[0][wave][0], lane);

        int buf = 0;
        for (int d0 = 0; d0 < Dd; d0 += CHD) {
            const bool more = (d0 + CHD) < Dd;
            if (more)
                stage_panel(w1base + (size_t)(d0 + CHD) * EHh, &s_b[buf ^ 1][wave][0], lane);
            wait_panel(more);                   // current panel resident in LDS

            const int seg = d0 >> 11;           // which omics segment (G = 2048)
            const float* xp = (seg == 0 ? x_rna : (seg == 1 ? x_cnv : x_met))
                              + (size_t)brow * Gg;
            const int gbase = d0 & (Gg - 1);
            const float* lb = &s_b[buf][wave][0];
#pragma unroll
            for (int r = 0; r < CHD; r += 4) {
                const int g = gbase + r;
                float m0 = gene_mask[(size_t)(g + ka)     * Pp + p];
                float m1 = gene_mask[(size_t)(g + ka + 1) * Pp + p];
                v2f a; a.x = xp[g + ka] * m0; a.y = xp[g + ka + 1] * m1;
                v2f b; b.x = lb[(r + ka) * 16 + lm];
                       b.y = lb[(r + ka + 1) * 16 + lm];
                c = wmma4(a, b, c);
            }
            buf ^= 1;
        }

        // bias + BatchNorm(eval) + ReLU, per hidden column
        const int ncol = wave * 16 + lm;
        const float bias  = eb1[p * EHh + ncol];
        const float scale = bn_g[p * EHh + ncol] * rsqrtf(bn_v[p * EHh + ncol] + BN_EPS);
        const float mn    = bn_m[p * EHh + ncol];
        const float bb    = bn_b[p * EHh + ncol];
#pragma unroll
        for (int j = 0; j < 8; ++j) {
            float h = (c[j] + bias - mn) * scale + bb;
            s_h[(j + 8 * half) * EHh + ncol] = fmaxf(h, 0.f);
        }
        __syncthreads();

        if (wave == 0) {
            // feat(16x16) = h(16x64) @ eW2[p](64x16), K=64 -> 16 WMMA steps
            v8f f = {};
#pragma unroll
            for (int k0 = 0; k0 < EHh; k0 += 4) {
                v2f a; a.x = s_h[lm * EHh + k0 + ka];
                       a.y = s_h[lm * EHh + k0 + ka + 1];
                v2f b; b.x = eW2[((size_t)p * EHh + k0 + ka)     * EOo + lm];
                       b.y = eW2[((size_t)p * EHh + k0 + ka + 1) * EOo + lm];
                f = wmma4(a, b, f);
            }
            const float be = eb2[p * EOo + lm];
#pragma unroll
            for (int j = 0; j < 8; ++j) {
                const int   mrow = j + 8 * half;
                const float fv   = f[j] + be;
                // padded rows have weight 0 -> contributes 0
                atomicAdd(&weighted[(size_t)s_rows[mrow] * EOo + lm], s_w[mrow] * fv);
            }
        }
        __syncthreads();
    }
}

// ---------------------------------------------------------------------------
// Kernel 4: classifier  relu(weighted @ cW1 + cb1) @ cW2 + cb2 -> [B, 1]
// ---------------------------------------------------------------------------
__global__ void k_cls(const float* __restrict__ weighted,
                      const float* __restrict__ cW1, const float* __restrict__ cb1,
                      const float* __restrict__ cW2, const float* __restrict__ cb2,
                      float* __restrict__ out) {
    const int b = blockIdx.x * blockDim.x + threadIdx.x;
    if (b >= Bsz) return;
    float w[EOo];
#pragma unroll
    for (int i = 0; i < EOo; ++i) w[i] = weighted[(size_t)b * EOo + i];
    float acc = cb2[0];
#pragma unroll
    for (int o = 0; o < CHh; ++o) {
        float s = cb1[o];
#pragma unroll
        for (int i = 0; i < EOo; ++i) s += w[i] * cW1[i * CHh + o];
        acc += fmaxf(s, 0.f) * cW2[o];
    }
    out[b] = acc;
}

// ---------------------------------------------------------------------------
extern "C" void kernel_launch(void* const* d_in, const int* in_sizes, int n_in,
                              void* d_out, int out_size, void* d_ws, size_t ws_size,
                              hipStream_t stream) {
    const float* x_rna   = (const float*)d_in[0];
    const float* x_cnv   = (const float*)d_in[1];
    const float* x_met   = (const float*)d_in[2];
    const float* g_mask  = (const float*)d_in[3];
    const float* gate_W1 = (const float*)d_in[4];
    const float* gate_b1 = (const float*)d_in[5];
    const float* gate_W2 = (const float*)d_in[6];
    const float* gate_b2 = (const float*)d_in[7];
    const float* eW1     = (const float*)d_in[8];
    const float* eb1     = (const float*)d_in[9];
    const float* bn_g    = (const float*)d_in[10];
    const float* bn_b    = (const float*)d_in[11];
    const float* bn_m    = (const float*)d_in[12];
    const float* bn_v    = (const float*)d_in[13];
    const float* eW2     = (const float*)d_in[14];
    const float* eb2     = (const float*)d_in[15];
    const float* cW1     = (const float*)d_in[16];
    const float* cb1     = (const float*)d_in[17];
    const float* cW2     = (const float*)d_in[18];
    const float* cb2     = (const float*)d_in[19];

    float* out_logits = (float*)d_out;                     // [B*NC]
    float* out_gw     = (float*)d_out + (size_t)Bsz * NCc; // [B*P]

    // Workspace layout (bytes)
    char* ws = (char*)d_ws;
    float* gh       = (float*)(ws);                         // B*GH floats
    float* weighted = (float*)(ws + 524288);                // B*EO floats
    int*   counts   = (int*)  (ws + 524288 + 65536);        // P ints
    int*   rows     = (int*)  (ws + 524288 + 65536 + 4096); // P*B ints
    float* wvals    = (float*)(ws + 524288 + 65536 + 4096 + 1048576); // P*B floats

    k_zero<<<dim3((Bsz * EOo + 255) / 256), dim3(256), 0, stream>>>(counts, weighted);

    k_gate_hidden<<<dim3(Bsz / 16, GHh / 16), dim3(32), 0, stream>>>(
        x_rna, gate_W1, gate_b1, gh);

    k_gate_topk<<<dim3(Bsz), dim3(Pp), 0, stream>>>(
        gh, gate_W2, gate_b2, out_gw, counts, rows, wvals);

    k_expert<<<dim3(Pp), dim3(128), 0, stream>>>(
        x_rna, x_cnv, x_met, g_mask, eW1, eb1, bn_g, bn_b, bn_m, bn_v,
        eW2, eb2, counts, rows, wvals, weighted);

    k_cls<<<dim3((Bsz + 255) / 256), dim3(256), 0, stream>>>(
        weighted, cW1, cb1, cW2, cb2, out_logits);
}